// Linear_85298050498602
// MI455X (gfx1250) — compile-verified
//
#include <hip/hip_runtime.h>
#include <math.h>

// K = sigma_v^2 * (x1-off)@(x2-off)^T + sigma_b^2
// N = M = 8192, P = 16. Output-write bound: 256MB / 23.3 TB/s ~ 11.5 us floor.
// Strategy: fp32 WMMA (V_WMMA_F32_16X16X4_F32 x4 per tile, exact fp32 math),
// scalars folded into A/B (x sigma_v) and C (= sigma_b^2), non-temporal stores.

typedef __attribute__((ext_vector_type(2))) float v2f;
typedef __attribute__((ext_vector_type(8))) float v8f;

#define N_DIM 8192
#define M_DIM 8192
#define P_DIM 16
#define MT_PER_WAVE 8   // m-tiles per wave (amortize A load + exp setup)

__global__ __launch_bounds__(256) void gp_linear_wmma_kernel(
    const float* __restrict__ x1,
    const float* __restrict__ x2,
    const float* __restrict__ p_log_sigma_b,
    const float* __restrict__ p_log_sigma_v,
    const float* __restrict__ p_offset,
    float* __restrict__ out)
{
    const int lane        = threadIdx.x & 31;
    const int waveInBlock = threadIdx.x >> 5;
    const int waveId      = blockIdx.x * (blockDim.x >> 5) + waveInBlock;

    const int mGroups = (M_DIM / 16) / MT_PER_WAVE;     // 64 groups of 8 m-tiles
    const int nTile   = waveId / mGroups;
    const int mGroup  = waveId - nTile * mGroups;
    const int n0      = nTile * 16;
    const int m0base  = mGroup * (16 * MT_PER_WAVE);

    // Scalars (single-element device arrays); tiny, L2/L0 resident.
    const float offset = p_offset[0];
    const float sv     = expf(p_log_sigma_v[0]);            // sigma_v
    const float sb2    = expf(2.0f * p_log_sigma_b[0]);     // sigma_b^2

    // Fragment addressing (ISA 7.12.2, 32-bit 16x4 A-matrix):
    //   lanes 0-15  : VGPR0=K0, VGPR1=K1   (rows M=lane)
    //   lanes 16-31 : VGPR0=K2, VGPR1=K3   (rows M=lane-16)
    // B 4x16 mirrors this with N=lane&15 across lanes.
    const int sub = lane >> 4;      // half-wave select
    const int r   = lane & 15;      // row (A: M) / col (B: N) within tile

    // ---- Load A fragments once: 4 chunks of 16x4 covering K = 0..15 ----
    v2f a[4];
    const float* aRow = x1 + (size_t)(n0 + r) * P_DIM + 2 * sub;
    #pragma unroll
    for (int k = 0; k < 4; ++k) {
        a[k].x = sv * (aRow[4 * k + 0] - offset);
        a[k].y = sv * (aRow[4 * k + 1] - offset);
    }

    #pragma unroll
    for (int mt = 0; mt < MT_PER_WAVE; ++mt) {
        const int m0 = m0base + mt * 16;

        // ---- Load B fragments: x2 rows are output columns ----
        v2f b[4];
        const float* bRow = x2 + (size_t)(m0 + r) * P_DIM + 2 * sub;
        #pragma unroll
        for (int k = 0; k < 4; ++k) {
            b[k].x = sv * (bRow[4 * k + 0] - offset);
            b[k].y = sv * (bRow[4 * k + 1] - offset);
        }

        // ---- Accumulator starts at sigma_b^2 so D is the final answer ----
        v8f c;
        #pragma unroll
        for (int i = 0; i < 8; ++i) c[i] = sb2;

        // 4 chained v_wmma_f32_16x16x4_f32: exact fp32 K=16 reduction
        #pragma unroll
        for (int k = 0; k < 4; ++k) {
            c = __builtin_amdgcn_wmma_f32_16x16x4_f32(
                /*neg_a=*/false, a[k], /*neg_b=*/false, b[k],
                /*c_mod=*/(short)0, c, /*reuse_a=*/false, /*reuse_b=*/false);
        }

        // ---- Store 16x16 f32 tile (C layout: VGPR rr -> rows rr / rr+8) ----
        // Write-once 256MB stream > 192MB L2: use non-temporal stores.
        float* o = out + (size_t)(n0 + 8 * sub) * M_DIM + (m0 + r);
        #pragma unroll
        for (int rr = 0; rr < 8; ++rr) {
            __builtin_nontemporal_store(c[rr], o + (size_t)rr * M_DIM);
        }
    }
}

extern "C" void kernel_launch(void* const* d_in, const int* in_sizes, int n_in,
                              void* d_out, int out_size, void* d_ws, size_t ws_size,
                              hipStream_t stream) {
    (void)in_sizes; (void)n_in; (void)out_size; (void)d_ws; (void)ws_size;
    const float* x1  = (const float*)d_in[0];
    const float* x2  = (const float*)d_in[1];
    const float* lsb = (const float*)d_in[2];
    const float* lsv = (const float*)d_in[3];
    const float* off = (const float*)d_in[4];
    float* out = (float*)d_out;

    const int totalWaves = (N_DIM / 16) * ((M_DIM / 16) / MT_PER_WAVE); // 32768
    const int wavesPerBlock = 256 / 32;                                 // 8
    const int blocks = totalWaves / wavesPerBlock;                      // 4096

    gp_linear_wmma_kernel<<<dim3(blocks), dim3(256), 0, stream>>>(
        x1, x2, lsb, lsv, off, out);
}